// E3nnModel_43078521979344
// MI455X (gfx1250) — compile-verified
//
#include <hip/hip_runtime.h>
#include <math.h>

#define BLOCK 256
#define K 4                           // nodes per thread
#define NODES_PER_BLOCK (BLOCK * K)   // 1024
#define NODES_PER_WAVE  (32 * K)      // 128

typedef int   v4i __attribute__((ext_vector_type(4)));
typedef float v4f __attribute__((ext_vector_type(4)));
typedef float v2f __attribute__((ext_vector_type(2)));

// ---------- CDNA5 helpers ----------

typedef __attribute__((address_space(3))) char lds_char_t;

__device__ __forceinline__ unsigned lds_byte_offset(const void* p) {
  return (unsigned)(unsigned long long)(lds_char_t*)p;
}

__device__ __forceinline__ void wait_asynccnt0() {
#if __has_builtin(__builtin_amdgcn_s_wait_asynccnt)
  __builtin_amdgcn_s_wait_asynccnt(0);
#else
  asm volatile("s_wait_asynccnt 0" ::: "memory");
#endif
}

// tanh via v_exp/v_rcp: ~6 VALU, saturates correctly at +/-inf.
__device__ __forceinline__ float fast_tanh(float x) {
  float e = __expf(2.0f * x);
  return 1.0f - 2.0f / (e + 1.0f);
}

__device__ __forceinline__ v2f vmax0(v2f a) {
  return __builtin_elementwise_max(a, (v2f)0.0f);
}
__device__ __forceinline__ v2f vfma(v2f a, v2f b, v2f c) {
  return __builtin_elementwise_fma(a, b, c);   // -> v_pk_fma_f32
}

// ---------- main kernel (first in file so --disasm snippet shows it) ----------

__global__ __launch_bounds__(BLOCK)
void e3nn_main(const int*   __restrict__ z,
               const float* __restrict__ pos,
               const int*   __restrict__ batch,
               const float* __restrict__ w_tp,
               const float* __restrict__ fc_w,
               const float* __restrict__ fc_b,
               float* __restrict__ sums,
               float* __restrict__ cnts,
               int n) {
  // Per-wave pos slab: 128 nodes * 12B = 1536B; 8 waves -> 12 KB.
  __shared__ __align__(16) float spos[3 * NODES_PER_BLOCK];

  const int tid  = threadIdx.x;
  const int lane = tid & 31;
  const int wave = tid >> 5;
  const int blockBase = blockIdx.x * NODES_PER_BLOCK;
  const int waveBase  = blockBase + wave * NODES_PER_WAVE;
  const int i0        = blockBase + tid * K;     // first node of this thread
  const bool fullBlock = (blockBase + NODES_PER_BLOCK <= n);

  // ---- Kick off the async pos DMA FIRST so it overlaps the weight loads
  //      and the z/batch streams. One base address pair; the 512B steps are
  //      encoded in IOFFSET, which the TDM adds to BOTH the LDS dest and the
  //      global source (ISA 15.18).
  if (fullBlock) {
    unsigned long long ga =
        (unsigned long long)((const char*)(pos + (size_t)waveBase * 3) + lane * 16);
    unsigned lbase =
        lds_byte_offset(&spos[wave * (3 * NODES_PER_WAVE)]) + lane * 16u;
    asm volatile("global_load_async_to_lds_b128 %0, %1, off"
                 :: "v"(lbase), "v"(ga) : "memory");
    asm volatile("global_load_async_to_lds_b128 %0, %1, off offset:512"
                 :: "v"(lbase), "v"(ga) : "memory");
    asm volatile("global_load_async_to_lds_b128 %0, %1, off offset:1024"
                 :: "v"(lbase), "v"(ga) : "memory");
  }

  const float INV_SQRT2 = 0.70710678118654752f;
  const float SQRT2     = 1.41421356237309515f;
  const float INV_SQRT3 = 0.57735026918962576f;
  const float INV_SQRT6 = 0.40824829046386302f;

  // Pre-folded weights (uniform addresses -> scalar loads, in DMA shadow).
  const float wA0h = w_tp[0] * INV_SQRT2, wA1h = w_tp[1] * INV_SQRT2;
  const float wB0h = w_tp[2] * INV_SQRT2, wB1h = w_tp[3] * INV_SQRT2;
  const float wCD0 = (w_tp[4] + w_tp[6]) * INV_SQRT2;
  const float wCD1 = (w_tp[5] + w_tp[7]) * INV_SQRT2;
  const float wE   = w_tp[8];
  const float wE2  = wE * SQRT2;      // f8..f10
  const float wEh  = wE * INV_SQRT2;  // f11
  const float wE6  = wE * INV_SQRT6;  // f12
  float fw[13];
  #pragma unroll
  for (int k = 0; k < 13; ++k) fw[k] = fc_w[k];
  const float fb = fc_b[0];

  // Evaluate TWO independent nodes per call in packed-f32 registers
  // (v_pk_fma_f32 / v_pk_mul_f32 on CDNA5 VOP3P).
  auto eval2 = [&](v2f s, v2f x, v2f y_, v2f zc) -> v2f {
    v2f xx = x * x, yy = y_ * y_, zz = zc * zc;
    v2f ss  = s * s;
    v2f txy = xx + yy;
    v2f dot = txy + zz;
    v2f d3  = dot * INV_SQRT3;
    v2f sv0 = s * x, sv1 = s * y_, sv2 = s * zc;

    v2f f0  = vfma(ss, (v2f)wA0h, d3 * wB0h);
    v2f f1  = vfma(ss, (v2f)wA1h, d3 * wB1h);
    v2f f2  = sv0 * wCD0, f3 = sv1 * wCD0, f4 = sv2 * wCD0;
    v2f f5  = sv0 * wCD1, f6 = sv1 * wCD1, f7 = sv2 * wCD1;
    v2f f8  = (x * y_) * wE2;
    v2f f9  = (y_ * zc) * wE2;
    v2f f10 = (x * zc) * wE2;
    v2f f11 = (xx - yy) * wEh;
    v2f f12 = vfma(zz, (v2f)2.0f, -txy) * wE6;

    v2f acc = (v2f)fb;
    acc = vfma(vmax0(f0 ), (v2f)fw[0] , acc);
    acc = vfma(vmax0(f1 ), (v2f)fw[1] , acc);
    acc = vfma(vmax0(f2 ), (v2f)fw[2] , acc);
    acc = vfma(vmax0(f3 ), (v2f)fw[3] , acc);
    acc = vfma(vmax0(f4 ), (v2f)fw[4] , acc);
    acc = vfma(vmax0(f5 ), (v2f)fw[5] , acc);
    acc = vfma(vmax0(f6 ), (v2f)fw[6] , acc);
    acc = vfma(vmax0(f7 ), (v2f)fw[7] , acc);
    acc = vfma(vmax0(f8 ), (v2f)fw[8] , acc);
    acc = vfma(vmax0(f9 ), (v2f)fw[9] , acc);
    acc = vfma(vmax0(f10), (v2f)fw[10], acc);
    acc = vfma(vmax0(f11), (v2f)fw[11], acc);
    acc = vfma(vmax0(f12), (v2f)fw[12], acc);

    v2f r;
    r.x = fast_tanh(acc.x);
    r.y = fast_tanh(acc.y);
    return r;
  };

  // Per-thread running segment (batch is sorted).
  int   gcur = -1;
  float acc  = 0.0f, cnt = 0.0f;

  if (fullBlock) {
    // Streamed-once scalars while the DMA is in flight (non-temporal B128s).
    v4i z4 = __builtin_nontemporal_load((const v4i*)(z + i0));
    v4i g4 = __builtin_nontemporal_load((const v4i*)(batch + i0));

    wait_asynccnt0();

    // 12 floats (4 nodes) per lane from LDS, 16B-aligned ds_load_b128 x3.
    const v4f* lp =
        (const v4f*)(spos + (size_t)(wave * (3 * NODES_PER_WAVE)) + lane * 12);
    v4f p0 = lp[0], p1 = lp[1], p2 = lp[2];

    // Pack node pairs {0,1} and {2,3} into v2f lanes.
    v2f sA = {(float)z4.x, (float)z4.y};
    v2f sB = {(float)z4.z, (float)z4.w};
    v2f xA = {p0.x, p0.w}, yA = {p0.y, p1.x}, zA = {p0.z, p1.y};
    v2f xB = {p1.z, p2.y}, yB = {p1.w, p2.z}, zB = {p2.x, p2.w};

    v2f yPA = eval2(sA, xA, yA, zA);
    v2f yPB = eval2(sB, xB, yB, zB);

    float ys[K] = {yPA.x, yPA.y, yPB.x, yPB.y};
    int   gs[K] = {g4.x, g4.y, g4.z, g4.w};

    gcur = gs[0]; acc = ys[0]; cnt = 1.0f;
    #pragma unroll
    for (int k = 1; k < K; ++k) {
      if (gs[k] == gcur) { acc += ys[k]; cnt += 1.0f; }
      else {
        // rare (avg run length = 64 nodes): flush mid-thread boundary
        atomicAdd(&sums[gcur], acc);
        atomicAdd(&cnts[gcur], cnt);
        gcur = gs[k]; acc = ys[k]; cnt = 1.0f;
      }
    }
  } else {
    // ---- tail path (not hit for N = 4M, kept for generality) ----
    for (int k = 0; k < K; ++k) {
      int i = i0 + k;
      if (i >= n) break;
      v2f s  = (v2f)((float)z[i]);
      v2f v0 = (v2f)pos[(size_t)i * 3 + 0];
      v2f v1 = (v2f)pos[(size_t)i * 3 + 1];
      v2f v2 = (v2f)pos[(size_t)i * 3 + 2];
      float y = eval2(s, v0, v1, v2).x;
      int   g = batch[i];
      if (g == gcur)      { acc += y; cnt += 1.0f; }
      else {
        if (gcur >= 0) { atomicAdd(&sums[gcur], acc); atomicAdd(&cnts[gcur], cnt); }
        gcur = g; acc = y; cnt = 1.0f;
      }
    }
  }

  // Wave32 segmented scan over per-thread tail runs (ds_bpermute shuffles);
  // only run-tail lanes issue global_atomic_add_f32 (no-return).
  #pragma unroll
  for (int off = 1; off < 32; off <<= 1) {
    float au = __shfl_up(acc,  off, 32);
    float cu = __shfl_up(cnt,  off, 32);
    int   gu = __shfl_up(gcur, off, 32);
    if (lane >= off && gu == gcur) { acc += au; cnt += cu; }
  }
  int g_next = __shfl_down(gcur, 1, 32);
  bool is_tail = (lane == 31) || (g_next != gcur);
  if (is_tail && gcur >= 0) {
    atomicAdd(&sums[gcur], acc);
    atomicAdd(&cnts[gcur], cnt);
  }
}

// ---------- small kernels ----------

__global__ void e3nn_zero(float* __restrict__ ws, int nfloats) {
  int i = blockIdx.x * blockDim.x + threadIdx.x;
  if (i < nfloats) ws[i] = 0.0f;
}

__global__ void e3nn_finalize(const float* __restrict__ sums,
                              const float* __restrict__ cnts,
                              float* __restrict__ out, int ngraphs) {
  int gidx = blockIdx.x * blockDim.x + threadIdx.x;
  if (gidx < ngraphs) out[gidx] = sums[gidx] / fmaxf(cnts[gidx], 1.0f);
}

// ---------- launcher ----------

extern "C" void kernel_launch(void* const* d_in, const int* in_sizes, int n_in,
                              void* d_out, int out_size, void* d_ws, size_t ws_size,
                              hipStream_t stream) {
  const int*   z     = (const int*)  d_in[0];
  const float* pos   = (const float*)d_in[1];
  const int*   batch = (const int*)  d_in[2];
  const float* w_tp  = (const float*)d_in[3];
  const float* fc_w  = (const float*)d_in[4];
  const float* fc_b  = (const float*)d_in[5];

  const int n       = in_sizes[0];   // 4,194,304 nodes
  const int ngraphs = out_size;      // 65,536 graphs

  float* sums = (float*)d_ws;
  float* cnts = sums + ngraphs;
  const int nacc = 2 * ngraphs;

  e3nn_zero<<<(nacc + BLOCK - 1) / BLOCK, BLOCK, 0, stream>>>(sums, nacc);
  e3nn_main<<<(n + NODES_PER_BLOCK - 1) / NODES_PER_BLOCK, BLOCK, 0, stream>>>(
      z, pos, batch, w_tp, fc_w, fc_b, sums, cnts, n);
  e3nn_finalize<<<(ngraphs + BLOCK - 1) / BLOCK, BLOCK, 0, stream>>>(
      sums, cnts, (float*)d_out, ngraphs);
}